// CLCRec_Graph_66537633350154
// MI455X (gfx1250) — compile-verified
//
#include <hip/hip_runtime.h>
#include <math.h>

typedef __attribute__((ext_vector_type(2))) float v2f;
typedef __attribute__((ext_vector_type(8))) float v8f;
typedef __attribute__((ext_vector_type(4))) int   v4i;

#define NUM_USER 100000
#define NUM_ITEM 50000
#define DIM_E 128
#define DIM_FEAT 256
#define BATCH 1024
#define NUM_NEG 127
#define FLAT (BATCH * (1 + NUM_NEG))   // 131072

// ---------------------------------------------------------------------------
// Async global->LDS staging (CDNA5 GLOBAL_LOAD_ASYNC_TO_LDS_B128, ASYNCcnt),
// with a safe fallback to plain LDS copies if the builtin is unavailable.
// Builtin signature (from clang diagnostic): (AS1 int4*, AS3 int4*, Ii, Ii).
// ---------------------------------------------------------------------------
#if __has_builtin(__builtin_amdgcn_global_load_async_to_lds_b128)
#define HAS_ASYNC_LDS 1
#else
#define HAS_ASYNC_LDS 0
#endif

#if HAS_ASYNC_LDS
typedef __attribute__((address_space(1))) v4i* gbl_v4i;
typedef __attribute__((address_space(3))) v4i* lds_v4i;
#endif

__device__ __forceinline__ void stage_tile(const float* __restrict__ gsrc,
                                           float* lds_dst, int tid, int nthreads) {
  // 16x256 f32 tile = 16 KB = 1024 x 16B, contiguous in both global and LDS.
#if HAS_ASYNC_LDS
  for (int i = tid; i < 1024; i += nthreads) {
    // generic global addr == AS(1) addr numerically
    gbl_v4i g = (gbl_v4i)(unsigned long long)(uintptr_t)(gsrc + i * 4);
    // low 32 bits of generic LDS addr == LDS byte offset (flat aperture rule)
    lds_v4i l = (lds_v4i)(unsigned int)(uintptr_t)(lds_dst + i * 4);
    __builtin_amdgcn_global_load_async_to_lds_b128(g, l, 0, 0);
  }
#else
  for (int i = tid; i < 1024; i += nthreads) {
    reinterpret_cast<float4*>(lds_dst)[i] = reinterpret_cast<const float4*>(gsrc)[i];
  }
#endif
}

__device__ __forceinline__ void wait_stage() {
#if HAS_ASYNC_LDS
#if __has_builtin(__builtin_amdgcn_s_wait_asynccnt)
  __builtin_amdgcn_s_wait_asynccnt(0);
#else
  asm volatile("s_wait_asynccnt 0x0" ::: "memory");
#endif
#endif
}

// ---------------------------------------------------------------------------
// Kernel T: pre-transpose weights so every 16-column B-tile becomes a
// contiguous 16 KB block ([n][k] layout) -> coalesced/async staging.
// ---------------------------------------------------------------------------
__global__ __launch_bounds__(256) void transpose_w_kernel(
    const float* __restrict__ W1, const float* __restrict__ W2,
    float* __restrict__ W1T, float* __restrict__ W2T) {
  const int i = blockIdx.x * blockDim.x + threadIdx.x;
  if (i < 256 * 256) {
    const int k = i >> 8, n = i & 255;
    W1T[n * 256 + k] = W1[k * 256 + n];
  }
  if (i < 256 * 128) {
    const int k = i >> 7, n = i & 127;
    W2T[n * 256 + k] = W2[k * 128 + n];
  }
}

// ---------------------------------------------------------------------------
// Kernel 1: feature = l2norm(v_feat) @ W1 + b1 -> leaky_relu -> @ W2 + b2
// 4 waves/block, 16 rows/wave (64 rows/block). fp32 WMMA 16x16x4.
// B tiles staged once per block into double-buffered LDS via async-to-LDS,
// overlapped with WMMA compute on the other buffer.
// ---------------------------------------------------------------------------
__global__ __launch_bounds__(128) void mlp_feature_kernel(
    const float* __restrict__ vfeat,   // [50000,256]
    const float* __restrict__ W1T,     // [256,256]  (transposed: [n][k])
    const float* __restrict__ b1,      // [256]
    const float* __restrict__ W2T,     // [128,256]  (transposed: [n][k])
    const float* __restrict__ b2,      // [128]
    float* __restrict__ feature)       // [50000,128]
{
  __shared__ float As[4 * 16 * 256];  // per-wave normalized input (64 KB)
  __shared__ float Hs[4 * 16 * 256];  // per-wave hidden acts      (64 KB)
  __shared__ float Bs[2 * 16 * 256];  // shared W tile, double buf (32 KB)

  const int tid  = threadIdx.x;
  const int lane = tid & 31;
  const int wave = tid >> 5;
  const int rowbase = blockIdx.x * 64 + wave * 16;

  float* As_w = As + wave * 4096;
  float* Hs_w = Hs + wave * 4096;

  // kick off async staging of B tile 0 while we load/normalize A
  stage_tile(W1T, Bs, tid, 128);

  // ---- load this wave's 16x256 v_feat tile (row-clamped for the tail) ----
  {
    const float4* vf4 = reinterpret_cast<const float4*>(vfeat);
    float4* dst = reinterpret_cast<float4*>(As_w);
    for (int idx = lane; idx < 16 * 64; idx += 32) {
      int grow = rowbase + (idx >> 6);
      if (grow > NUM_ITEM - 1) grow = NUM_ITEM - 1;
      dst[idx] = vf4[(size_t)grow * 64 + (idx & 63)];
    }
  }
  // ---- l2 normalize each row: lane (r, half) handles 128 elements ----
  {
    const int r = lane & 15, h = lane >> 4;
    float* ap = As_w + r * 256 + h * 128;
    float s = 0.f;
    for (int j = 0; j < 128; ++j) s += ap[j] * ap[j];
    s += __shfl_xor(s, 16, 32);
    const float scale = 1.0f / fmaxf(sqrtf(s), 1e-12f);
    for (int j = 0; j < 128; ++j) ap[j] *= scale;
  }

  wait_stage();
  __syncthreads();  // B tile 0 visible to all waves

  const int m16 = lane & 15;   // M for A / N for B / N for C
  const int g   = lane >> 4;   // lane group

  // unified tile loop: tiles 0..15 = GEMM1 (-> Hs), tiles 16..23 = GEMM2 (-> out)
  for (int nt = 0; nt < 24; ++nt) {
    // prefetch next B tile into the other buffer (weights only, no dependency)
    if (nt + 1 < 24) {
      const float* nsrc = (nt + 1 < 16) ? (W1T + (size_t)(nt + 1) * 4096)
                                        : (W2T + (size_t)(nt + 1 - 16) * 4096);
      stage_tile(nsrc, Bs + ((nt + 1) & 1) * 4096, tid, 128);
    }

    const float* cur = Bs + (nt & 1) * 4096;
    const float* A   = (nt < 16) ? As_w : Hs_w;
    const float* arow = A   + m16 * 256 + g * 2;
    const float* brow = cur + m16 * 256 + g * 2;

    v8f c = {};
    for (int kk = 0; kk < 64; ++kk) {
      v2f a = *(const v2f*)(arow + kk * 4);
      v2f b = *(const v2f*)(brow + kk * 4);
      c = __builtin_amdgcn_wmma_f32_16x16x4_f32(false, a, false, b,
                                                (short)0, c, false, false);
    }

    // C layout: VGPR r -> M = r + 8*g, N = m16
    if (nt < 16) {
      const float bias = b1[nt * 16 + m16];
      for (int r = 0; r < 8; ++r) {
        float h = c[r] + bias;
        h = h > 0.f ? h : 0.01f * h;  // leaky_relu slope 0.01
        Hs_w[(r + 8 * g) * 256 + nt * 16 + m16] = h;
      }
    } else {
      const int col = (nt - 16) * 16 + m16;
      const float bias = b2[col];
      for (int r = 0; r < 8; ++r) {
        const int row = rowbase + r + 8 * g;
        if (row < NUM_ITEM) feature[(size_t)row * 128 + col] = c[r] + bias;
      }
    }

    wait_stage();
    __syncthreads();  // next tile staged & everyone done with 'cur'
  }
}

// ---------------------------------------------------------------------------
// Kernel 0a/0b: replacement mask
// ---------------------------------------------------------------------------
__global__ __launch_bounds__(256) void clear_mask_kernel(float* __restrict__ mask, int n) {
  int i = blockIdx.x * blockDim.x + threadIdx.x;
  if (i < n) mask[i] = 0.f;
}

__global__ __launch_bounds__(256) void set_mask_kernel(const int* __restrict__ ridx,
                                                       float* __restrict__ mask, int n) {
  int i = blockIdx.x * blockDim.x + threadIdx.x;
  if (i < n) mask[ridx[i]] = 1.0f;  // duplicates write identical value -> safe
}

// ---------------------------------------------------------------------------
// Kernel 2: gathers + per-position dots/norms. One wave per flat position.
// ---------------------------------------------------------------------------
__global__ __launch_bounds__(256) void gather_score_kernel(
    const int* __restrict__ user,      // [FLAT]
    const int* __restrict__ item,      // [FLAT]
    const float* __restrict__ emb,     // [150000,128]
    const float* __restrict__ feat,    // [50000,128]
    const float* __restrict__ mask,    // [FLAT]
    float* __restrict__ score1, float* __restrict__ score2,
    float* __restrict__ normu,  float* __restrict__ normi)
{
  const int p = (blockIdx.x * blockDim.x + threadIdx.x) >> 5;  // flat position
  const int lane = threadIdx.x & 31;
  const int row = p >> 7;

  const int u  = user[p];
  const int it = item[p];
  const int ps = item[row << 7];  // positive item of this batch row

  const float4 eu = reinterpret_cast<const float4*>(emb  + (size_t)u  * 128)[lane];
  const float4 ei = reinterpret_cast<const float4*>(emb  + (size_t)it * 128)[lane];
  const float4 ep = reinterpret_cast<const float4*>(emb  + (size_t)ps * 128)[lane];
  const float4 ft = reinterpret_cast<const float4*>(feat + (size_t)(it - NUM_USER) * 128)[lane];

  float fv[4] = {ft.x, ft.y, ft.z, ft.w};
  float pv[4] = {ep.x, ep.y, ep.z, ep.w};
  float uv[4] = {eu.x, eu.y, eu.z, eu.w};
  float iv[4] = {ei.x, ei.y, ei.z, ei.w};

  float sf = 0.f, sp = 0.f, dfp = 0.f, de = 0.f, df = 0.f, su = 0.f, si = 0.f;
  for (int c = 0; c < 4; ++c) {
    sf  += fv[c] * fv[c];
    sp  += pv[c] * pv[c];
    dfp += fv[c] * pv[c];
    de  += uv[c] * iv[c];
    df  += uv[c] * fv[c];
    su  += uv[c] * uv[c];
    si  += iv[c] * iv[c];
  }
  for (int off = 16; off > 0; off >>= 1) {
    sf  += __shfl_xor(sf,  off, 32);
    sp  += __shfl_xor(sp,  off, 32);
    dfp += __shfl_xor(dfp, off, 32);
    de  += __shfl_xor(de,  off, 32);
    df  += __shfl_xor(df,  off, 32);
    su  += __shfl_xor(su,  off, 32);
    si  += __shfl_xor(si,  off, 32);
  }
  if (lane == 0) {
    const float nf = fmaxf(sqrtf(sf), 1e-12f);
    const float np = fmaxf(sqrtf(sp), 1e-12f);
    const float dot1 = dfp / (nf * np);                 // l2norm(pos) . l2norm(feat)
    const float dot2 = (mask[p] != 0.f) ? df : de;      // replaced rows use raw feat
    score1[p] = expf(dot1 * 0.5f);                      // TEMP = 2.0
    score2[p] = expf(dot2 * 0.5f);
    normu[p]  = sqrtf(su);
    normi[p]  = sqrtf(si);
  }
}

// ---------------------------------------------------------------------------
// Kernel 3a: per-batch-row softmax-style loss term
// ---------------------------------------------------------------------------
__global__ __launch_bounds__(128) void row_reduce_kernel(
    const float* __restrict__ score1, const float* __restrict__ score2,
    float* __restrict__ row1, float* __restrict__ row2)
{
  __shared__ float s1[128], s2[128];
  const int r = blockIdx.x, t = threadIdx.x;
  s1[t] = score1[r * 128 + t];
  s2[t] = score2[r * 128 + t];
  __syncthreads();
  for (int off = 64; off > 0; off >>= 1) {
    if (t < off) { s1[t] += s1[t + off]; s2[t] += s2[t + off]; }
    __syncthreads();
  }
  if (t == 0) {
    row1[r] = logf(s1[0]) - logf(score1[r * 128]);  // -log(pos/tot)
    row2[r] = logf(s2[0]) - logf(score2[r * 128]);
  }
}

// ---------------------------------------------------------------------------
// Kernel 3b: final deterministic reduction -> (total, reg_loss)
// ---------------------------------------------------------------------------
__global__ __launch_bounds__(256) void finalize_kernel(
    const float* __restrict__ row1, const float* __restrict__ row2,
    const float* __restrict__ normu, const float* __restrict__ normi,
    float* __restrict__ out)
{
  __shared__ float a[256], b[256], cu[256], ci[256];
  const int t = threadIdx.x;
  float s1 = 0.f, s2 = 0.f, su = 0.f, si = 0.f;
  for (int i = t; i < BATCH; i += 256) { s1 += row1[i]; s2 += row2[i]; }
  for (int i = t; i < FLAT; i += 256)  { su += normu[i]; si += normi[i]; }
  a[t] = s1; b[t] = s2; cu[t] = su; ci[t] = si;
  __syncthreads();
  for (int off = 128; off > 0; off >>= 1) {
    if (t < off) { a[t] += a[t+off]; b[t] += b[t+off]; cu[t] += cu[t+off]; ci[t] += ci[t+off]; }
    __syncthreads();
  }
  if (t == 0) {
    const float c1 = a[0] / (float)BATCH;
    const float c2 = b[0] / (float)BATCH;
    out[0] = 0.5f * c1 + 0.5f * c2;                               // LR_LAMBDA = 0.5
    out[1] = 0.5f * (cu[0] / (float)FLAT + ci[0] / (float)FLAT);  // reg_loss
  }
}

// ---------------------------------------------------------------------------
extern "C" void kernel_launch(void* const* d_in, const int* in_sizes, int n_in,
                              void* d_out, int out_size, void* d_ws, size_t ws_size,
                              hipStream_t stream) {
  const int*   user  = (const int*)d_in[0];
  const int*   item  = (const int*)d_in[1];
  const int*   ridx  = (const int*)d_in[2];
  const float* emb   = (const float*)d_in[3];
  const float* vfeat = (const float*)d_in[4];
  const float* W1    = (const float*)d_in[5];
  const float* b1    = (const float*)d_in[6];
  const float* W2    = (const float*)d_in[7];
  const float* b2    = (const float*)d_in[8];
  float* out = (float*)d_out;
  const int n_ridx = in_sizes[2];  // 65536

  // workspace layout (floats)
  float* ws      = (float*)d_ws;
  float* feature = ws;                          // 50000*128 = 6,400,000
  float* mask    = feature + (size_t)NUM_ITEM * DIM_E;
  float* score1  = mask   + FLAT;
  float* score2  = score1 + FLAT;
  float* normu   = score2 + FLAT;
  float* normi   = normu  + FLAT;
  float* row1    = normi  + FLAT;
  float* row2    = row1   + BATCH;
  float* W1T     = row2   + BATCH;              // 256*256
  float* W2T     = W1T    + 256 * 256;          // 128*256

  // 0) one-time weight transpose -> contiguous B tiles
  transpose_w_kernel<<<256, 256, 0, stream>>>(W1, W2, W1T, W2T);

  // 1) MLP item features via fp32 WMMA (782 blocks * 4 waves * 16 rows)
  mlp_feature_kernel<<<(NUM_ITEM + 63) / 64, 128, 0, stream>>>(vfeat, W1T, b1, W2T, b2, feature);

  // 2) replacement mask
  clear_mask_kernel<<<(FLAT + 255) / 256, 256, 0, stream>>>(mask, FLAT);
  set_mask_kernel<<<(n_ridx + 255) / 256, 256, 0, stream>>>(ridx, mask, n_ridx);

  // 3) gathers + per-position scores (one wave per position, 8 waves/block)
  gather_score_kernel<<<FLAT / 8, 256, 0, stream>>>(user, item, emb, feature, mask,
                                                    score1, score2, normu, normi);

  // 4) per-row contrastive terms
  row_reduce_kernel<<<BATCH, 128, 0, stream>>>(score1, score2, row1, row2);

  // 5) final scalars
  finalize_kernel<<<1, 256, 0, stream>>>(row1, row2, normu, normi, out);
}